// Binarization_60430189855367
// MI455X (gfx1250) — compile-verified
//
#include <hip/hip_runtime.h>
#include <stdint.h>

// high_order_residual (order=2), x: (11008, 4096) f32, mask: bool (u8).
// 1 block = 1 row; 256 threads (8 wave32). Row staged global->LDS via CDNA5
// async-to-LDS (ASYNCcnt, non-temporal), register-resident afterwards;
// 3 block reductions (shfl_xor wave32 + LDS cross-wave combine).

#define IC       4096
#define THREADS  256
#define EPT      16      // elements per thread
#define CPT      4       // float4 chunks per thread

typedef __attribute__((ext_vector_type(4))) float v4f;

__device__ __forceinline__ uint32_t lds_byte_off(const void* p) {
  // Low 32 bits of a generic LDS pointer == wave-relative LDS byte offset.
  return (uint32_t)(uintptr_t)p;
}

__device__ __forceinline__ void async_g2l_b128(uint32_t lds_off, const void* g) {
  // CDNA5: per-lane 16B copy global -> LDS, ASYNCcnt path, streaming (NT).
  asm volatile("global_load_async_to_lds_b128 %0, %1, off th:TH_LOAD_NT"
               :: "v"(lds_off), "v"((unsigned long long)(uintptr_t)g)
               : "memory");
}

__device__ __forceinline__ void wait_async0() {
  asm volatile("s_wait_asynccnt 0" ::: "memory");
}

__device__ __forceinline__ float wave_sum(float v) {
#pragma unroll
  for (int o = 16; o > 0; o >>= 1) v += __shfl_xor(v, o, 32);
  return v;
}

// sign(c)*scale with sign(0) == 0 (scale >= 0): v_bfi + v_cmp + v_cndmask
__device__ __forceinline__ float sgn_scale(float c, float scale) {
  const float t = __builtin_copysignf(scale, c);
  return (c != 0.0f) ? t : 0.0f;
}

__global__ __launch_bounds__(THREADS)
void hor2_kernel(const float* __restrict__ x,
                 const uint8_t* __restrict__ mask,
                 float* __restrict__ out) {
  __shared__ __align__(16) float   sx[IC];   // 16 KB
  __shared__ __align__(16) uint8_t smk[IC];  //  4 KB
  __shared__ float2 r1[8];
  __shared__ float4 r2[8];
  __shared__ float  r3[8];

  const int t    = threadIdx.x;
  const int wave = t >> 5;
  const int lane = t & 31;
  const int row  = blockIdx.x;

  const float*   gx = x    + (size_t)row * IC;
  const uint8_t* gm = mask + (size_t)row * IC;
  float*         go = out  + (size_t)row * IC;

  // ---------- async stage row into LDS ----------
  const uint32_t sx_base = lds_byte_off(&sx[0]);
  const uint32_t sm_base = lds_byte_off(&smk[0]);
#pragma unroll
  for (int k = 0; k < CPT; ++k) {
    const int ci = k * THREADS + t;                    // coalesced 16B chunks
    async_g2l_b128(sx_base + (uint32_t)ci * 16u, gx + ci * 4);
  }
  async_g2l_b128(sm_base + (uint32_t)t * 16u, gm + t * 16);   // 256*16B = 4 KB
  wait_async0();
  __syncthreads();

  // ---------- P1: LDS -> regs, local sum & count ----------
  float    c[EPT];          // holds v = x*m now, c = (v-mean1)*m after P2
  uint32_t mbits = 0;
  float    lsum  = 0.0f;
#pragma unroll
  for (int k = 0; k < CPT; ++k) {
    const int ci = k * THREADS + t;
    const v4f      xv = reinterpret_cast<const v4f*>(sx)[ci];
    const uint32_t mw = reinterpret_cast<const uint32_t*>(smk)[ci];
#pragma unroll
    for (int j = 0; j < 4; ++j) {
      const uint32_t mj = (mw >> (8 * j)) & 1u;
      const float    vi = mj ? xv[j] : 0.0f;
      c[k * 4 + j] = vi;
      lsum += vi;
      mbits |= mj << (k * 4 + j);
    }
  }
  {
    const float a = wave_sum(lsum);
    const float b = wave_sum((float)__popc(mbits));
    if (lane == 0) r1[wave] = make_float2(a, b);
  }
  __syncthreads();
  float row_sum = 0.0f, row_cnt = 0.0f;
#pragma unroll
  for (int w = 0; w < 8; ++w) { const float2 p = r1[w]; row_sum += p.x; row_cnt += p.y; }
  const float safe  = fmaxf(row_cnt, 1.0f);
  const float mean1 = row_sum / safe;                  // cnt==0 -> 0

  // ---------- P2: c = (v-mean1)*m ; sum(c), sum(|c|), sum(sign(c)) ----------
  float sc = 0.0f, sabs = 0.0f;
  int   npos = 0, nneg = 0;
#pragma unroll
  for (int i = 0; i < EPT; ++i) {
    const float mi = (float)((mbits >> i) & 1u);
    const float ci = (c[i] - mean1) * mi;              // exactly 0 if unmasked
    c[i] = ci;
    sc   += ci;
    sabs += fabsf(ci);
    npos += (int)__popcll(__ballot(ci > 0.0f));        // wave-reduced already
    nneg += (int)__popcll(__ballot(ci < 0.0f));
  }
  {
    const float a = wave_sum(sc);
    const float b = wave_sum(sabs);
    if (lane == 0) r2[wave] = make_float4(a, b, (float)(npos - nneg), 0.0f);
  }
  __syncthreads();
  float rsc = 0.0f, rsabs = 0.0f, rssg = 0.0f;
#pragma unroll
  for (int w = 0; w < 8; ++w) { const float4 p = r2[w]; rsc += p.x; rsabs += p.y; rssg += p.z; }
  const float scale1 = rsabs / safe;
  // residual2 = c - sign(c)*scale1  =>  mean2 = (sum(c) - scale1*sum(sign))/cnt
  const float mean2  = (rsc - scale1 * rssg) / safe;

  // ---------- P3: c2 = (c - sign(c)*scale1 - mean2)*m ; sum(|c2|) ----------
  float c2[EPT];
  float sabs2 = 0.0f;
#pragma unroll
  for (int i = 0; i < EPT; ++i) {
    const float mi = (float)((mbits >> i) & 1u);
    const float v2 = (c[i] - sgn_scale(c[i], scale1) - mean2) * mi;
    c2[i] = v2;
    sabs2 += fabsf(v2);
  }
  {
    const float a = wave_sum(sabs2);
    if (lane == 0) r3[wave] = a;
  }
  __syncthreads();
  float rsabs2 = 0.0f;
#pragma unroll
  for (int w = 0; w < 8; ++w) rsabs2 += r3[w];
  const float scale2 = rsabs2 / safe;
  const float m12    = mean1 + mean2;

  // ---------- P4: out = m * (sign(c)*scale1 + sign(c2)*scale2 + mean1+mean2) ----------
#pragma unroll
  for (int k = 0; k < CPT; ++k) {
    const int ci = k * THREADS + t;
    v4f o;
#pragma unroll
    for (int j = 0; j < 4; ++j) {
      const int   i  = k * 4 + j;
      const float mi = (float)((mbits >> i) & 1u);
      o[j] = mi * (sgn_scale(c[i], scale1) + sgn_scale(c2[i], scale2) + m12);
    }
    __builtin_nontemporal_store(o, reinterpret_cast<v4f*>(go) + ci);
  }
}

extern "C" void kernel_launch(void* const* d_in, const int* in_sizes, int n_in,
                              void* d_out, int out_size, void* d_ws, size_t ws_size,
                              hipStream_t stream) {
  const float*   x    = (const float*)d_in[0];
  const uint8_t* mask = (const uint8_t*)d_in[1];   // jnp.bool_ -> 1 byte/elem
  float*         out  = (float*)d_out;
  const int rows = in_sizes[0] / IC;               // 11008
  hipLaunchKernelGGL(hor2_kernel, dim3(rows), dim3(THREADS), 0, stream,
                     x, mask, out);
}